// MultiLatentAttention_35802847380107
// MI455X (gfx1250) — compile-verified
//
#include <hip/hip_runtime.h>
#include <hip/hip_bf16.h>
#include <math.h>

typedef __attribute__((ext_vector_type(2))) float v2f;
typedef __attribute__((ext_vector_type(8))) float v8f;

#define E_DIM 1024
#define H_DIM 16
#define P_DIM 64
#define B_DIM 2
#define S_DIM 2048

// ---------------- lambda = exp(q1.k1) - exp(q2.k2) + lambda_init ----------------
__global__ void mla_lam_kernel(const float* __restrict__ q1, const float* __restrict__ k1,
                               const float* __restrict__ q2, const float* __restrict__ k2,
                               const float* __restrict__ lam_init, float* __restrict__ lam_out) {
    __shared__ float s1[256];
    __shared__ float s2[256];
    int t = threadIdx.x;
    float a1 = 0.f, a2 = 0.f;
    for (int i = t; i < E_DIM; i += 256) { a1 += q1[i] * k1[i]; a2 += q2[i] * k2[i]; }
    s1[t] = a1; s2[t] = a2;
    __syncthreads();
    for (int w = 128; w > 0; w >>= 1) {
        if (t < w) { s1[t] += s1[t + w]; s2[t] += s2[t + w]; }
        __syncthreads();
    }
    if (t == 0) *lam_out = expf(s1[0]) - expf(s2[0]) + lam_init[0];
}

// ---------------- Reff[p,e] = sum_i (i+1) * RW[i*P + p, e]  ([64,1024]) ----------------
__global__ void mla_reff_kernel(const float* __restrict__ RW, float* __restrict__ Reff) {
    int idx = blockIdx.x * blockDim.x + threadIdx.x;   // 64*1024 threads
    int p = idx >> 10;
    int e = idx & 1023;
    float s = 0.f;
#pragma unroll
    for (int i = 0; i < H_DIM; ++i) s += (float)(i + 1) * RW[(i * P_DIM + p) * E_DIM + e];
    Reff[p * E_DIM + e] = s;
}

// ---------------- scale columns [32..63] of k (in place) by -lam ----------------
__global__ void mla_kscale_kernel(float* __restrict__ k, const float* __restrict__ lam) {
    int idx = blockIdx.x * blockDim.x + threadIdx.x;   // B*S*32 threads
    int row = idx >> 5;
    int col = 32 + (idx & 31);
    k[row * P_DIM + col] *= -(*lam);
}

// ---------------- wave-per-16x64-tile fp32 WMMA GEMM (static shapes) ----------------
// C[tm:tm+16, tn:tn+64] = alpha * A[16,K] x B[K,64] (+ bias)
// TRANSB=false: B row-major [K,N], LDB = N-stride
// TRANSB=true : B given as [N,K] row-major (logical B = givenT), LDB = K-stride
template <int K, int LDA, int LDB, int LDC, bool TRANSB>
__global__ __launch_bounds__(32)
void mla_gemm_wmma(const float* __restrict__ A, const float* __restrict__ B,
                   float* __restrict__ C, const float* __restrict__ bias, float alpha) {
    const int lane = threadIdx.x;          // one wave32 per block
    const int half = lane >> 4;            // selects K=0,1 vs K=2,3 within a 4-step
    const int mr   = lane & 15;            // M for A operand, N for B/C operands
    const int tn   = blockIdx.x * 64;
    const int tm   = blockIdx.y * 16;

    v8f acc0 = {}, acc1 = {}, acc2 = {}, acc3 = {};

    const float* aptr = A + (long)(tm + mr) * LDA + 2 * half;
    const float* bptr = TRANSB ? (B + (long)(tn + mr) * LDB + 2 * half)
                               : (B + (long)(2 * half) * LDB + tn + mr);

    for (int k0 = 0; k0 < K; k0 += 16) {
        if (K > 64) {
            __builtin_prefetch(aptr + 16, 0, 1);   // global_prefetch_b8 on next A chunk
        }
#pragma unroll
        for (int kk = 0; kk < 16; kk += 4) {
            v2f av = *(const v2f*)(aptr + kk);
            v2f b0, b1, b2, b3;
            if (TRANSB) {
                b0 = *(const v2f*)(bptr + kk + 0 * 16 * LDB);
                b1 = *(const v2f*)(bptr + kk + 1 * 16 * LDB);
                b2 = *(const v2f*)(bptr + kk + 2 * 16 * LDB);
                b3 = *(const v2f*)(bptr + kk + 3 * 16 * LDB);
            } else {
                b0.x = bptr[(kk + 0) * LDB +  0];  b0.y = bptr[(kk + 1) * LDB +  0];
                b1.x = bptr[(kk + 0) * LDB + 16];  b1.y = bptr[(kk + 1) * LDB + 16];
                b2.x = bptr[(kk + 0) * LDB + 32];  b2.y = bptr[(kk + 1) * LDB + 32];
                b3.x = bptr[(kk + 0) * LDB + 48];  b3.y = bptr[(kk + 1) * LDB + 48];
            }
            acc0 = __builtin_amdgcn_wmma_f32_16x16x4_f32(false, av, false, b0, (short)0, acc0, false, false);
            acc1 = __builtin_amdgcn_wmma_f32_16x16x4_f32(false, av, false, b1, (short)0, acc1, false, false);
            acc2 = __builtin_amdgcn_wmma_f32_16x16x4_f32(false, av, false, b2, (short)0, acc2, false, false);
            acc3 = __builtin_amdgcn_wmma_f32_16x16x4_f32(false, av, false, b3, (short)0, acc3, false, false);
        }
        aptr += 16;
        bptr += TRANSB ? 16 : 16 * LDB;
    }

    const float bb0 = bias ? bias[tn + mr +  0] : 0.f;
    const float bb1 = bias ? bias[tn + mr + 16] : 0.f;
    const float bb2 = bias ? bias[tn + mr + 32] : 0.f;
    const float bb3 = bias ? bias[tn + mr + 48] : 0.f;
    float* crow = C + (long)(tm + 8 * half) * LDC + tn + mr;
#pragma unroll
    for (int j = 0; j < 8; ++j) {
        // C/D layout: VGPR j -> M = j + 8*half, N = lane&15
        crow[(long)j * LDC +  0] = acc0[j] * alpha + bb0;
        crow[(long)j * LDC + 16] = acc1[j] * alpha + bb1;
        crow[(long)j * LDC + 32] = acc2[j] * alpha + bb2;
        crow[(long)j * LDC + 48] = acc3[j] * alpha + bb3;
    }
}

extern "C" void kernel_launch(void* const* d_in, const int* in_sizes, int n_in,
                              void* d_out, int out_size, void* d_ws, size_t ws_size,
                              hipStream_t stream) {
    const float* x        = (const float*)d_in[0];   // [B,S,E]
    const float* WQ       = (const float*)d_in[1];   // [E,E]
    const float* WK       = (const float*)d_in[2];
    const float* WV       = (const float*)d_in[3];
    const float* RW       = (const float*)d_in[4];   // result_weight [E,E]
    const float* proj_w   = (const float*)d_in[5];   // [E,P]
    const float* proj_b   = (const float*)d_in[6];   // [P]
    const float* q1v      = (const float*)d_in[7];
    const float* k1v      = (const float*)d_in[8];
    const float* q2v      = (const float*)d_in[9];
    const float* k2v      = (const float*)d_in[10];
    const float* lam_init = (const float*)d_in[11];
    float* out = (float*)d_out;

    // workspace layout (floats), total ~22 MB
    float* f     = (float*)d_ws;
    float* lam   = f;                           // 1 (padded to 16)
    float* WeffQ = f + 16;                      // 1024*64
    float* WeffK = WeffQ + E_DIM * P_DIM;
    float* WeffV = WeffK + E_DIM * P_DIM;
    float* Reff  = WeffV + E_DIM * P_DIM;       // 64*1024
    float* q     = Reff  + P_DIM * E_DIM;       // 4096*64
    float* k     = q + (long)B_DIM * S_DIM * P_DIM;
    float* v     = k + (long)B_DIM * S_DIM * P_DIM;
    float* base  = v + (long)B_DIM * S_DIM * P_DIM;
    float* Ascr  = base + (long)B_DIM * S_DIM * P_DIM;   // [S,S] reused per batch (16 MB)

    const dim3 wave(32, 1, 1);
    const float scale = 0.125f;   // 1/sqrt(E/H) = 1/8

    // 1) scalar lambda
    mla_lam_kernel<<<1, 256, 0, stream>>>(q1v, k1v, q2v, k2v, lam_init, lam);

    // 2) Reff = head-weighted fold of result_weight  [64,1024]
    mla_reff_kernel<<<(P_DIM * E_DIM) / 256, 256, 0, stream>>>(RW, Reff);

    // 3) Weff = W @ proj_w  ([1024,1024]x[1024,64])
    {
        dim3 g(1, E_DIM / 16, 1);
        mla_gemm_wmma<E_DIM, E_DIM, P_DIM, P_DIM, false><<<g, wave, 0, stream>>>(WQ, proj_w, WeffQ, nullptr, 1.0f);
        mla_gemm_wmma<E_DIM, E_DIM, P_DIM, P_DIM, false><<<g, wave, 0, stream>>>(WK, proj_w, WeffK, nullptr, 1.0f);
        mla_gemm_wmma<E_DIM, E_DIM, P_DIM, P_DIM, false><<<g, wave, 0, stream>>>(WV, proj_w, WeffV, nullptr, 1.0f);
    }

    // 4) q/k/v = x2d @ Weff + proj_b   ([4096,1024]x[1024,64])
    {
        dim3 g(1, (B_DIM * S_DIM) / 16, 1);
        mla_gemm_wmma<E_DIM, E_DIM, P_DIM, P_DIM, false><<<g, wave, 0, stream>>>(x, WeffQ, q, proj_b, 1.0f);
        mla_gemm_wmma<E_DIM, E_DIM, P_DIM, P_DIM, false><<<g, wave, 0, stream>>>(x, WeffK, k, proj_b, 1.0f);
        mla_gemm_wmma<E_DIM, E_DIM, P_DIM, P_DIM, false><<<g, wave, 0, stream>>>(x, WeffV, v, proj_b, 1.0f);
    }

    // 5) k' : scale second half of k columns by -lambda (in place)
    mla_kscale_kernel<<<(B_DIM * S_DIM * 32) / 256, 256, 0, stream>>>(k, lam);

    // 6) per batch: scores = scale * (q_b @ k'_bT), then base_b = scores @ v_b
    for (int b = 0; b < B_DIM; ++b) {
        const float* qb = q + (long)b * S_DIM * P_DIM;
        const float* kb = k + (long)b * S_DIM * P_DIM;
        const float* vb = v + (long)b * S_DIM * P_DIM;
        float* baseb    = base + (long)b * S_DIM * P_DIM;

        dim3 g1(S_DIM / 64, S_DIM / 16, 1);
        mla_gemm_wmma<P_DIM, P_DIM, P_DIM, S_DIM, true><<<g1, wave, 0, stream>>>(qb, kb, Ascr, nullptr, scale);

        dim3 g2(1, S_DIM / 16, 1);
        mla_gemm_wmma<S_DIM, S_DIM, P_DIM, P_DIM, false><<<g2, wave, 0, stream>>>(Ascr, vb, baseb, nullptr, 1.0f);
    }

    // 7) out = base @ Reff  ([4096,64]x[64,1024])
    {
        dim3 g(E_DIM / 64, (B_DIM * S_DIM) / 16, 1);
        mla_gemm_wmma<P_DIM, P_DIM, E_DIM, E_DIM, false><<<g, wave, 0, stream>>>(base, Reff, out, nullptr, 1.0f);
    }
}